// PositionalEncoding_23905787970124
// MI455X (gfx1250) — compile-verified
//
#include <hip/hip_runtime.h>
#include <math.h>

// PositionalEncoding: out[b,c,h,w,t] = x[b,c,h,w,t] + pe[c,t]
// x: [4, 64, 128, 128, 20] fp32.  Pure HBM-bandwidth-bound broadcast add.
// ~671 MB traffic -> ~29 us floor at 23.3 TB/s. No matrix ops: WMMA N/A.

typedef float v4f __attribute__((ext_vector_type(4)));  // native vec -> b128 + NT ok

namespace {
constexpr int kB          = 4;
constexpr int kC          = 64;
constexpr int kW          = 128;
constexpr int kT          = 20;
constexpr int kSlabElems  = kW * kW * kT;          // 327680 elems per (b,c) slab
constexpr int kSlabVec4   = kSlabElems / 4;        // 81920 float4 per slab
constexpr int kThreads    = 256;                   // 8 wave32 per block
constexpr int kVecPerThr  = 4;                     // 4 float4 (64B) per thread
constexpr int kBlocksX    = kSlabVec4 / (kThreads * kVecPerThr); // 80
static_assert(kBlocksX * kThreads * kVecPerThr == kSlabVec4, "exact tiling");
}

__global__ __launch_bounds__(kThreads)
void pe_add_kernel(const float* __restrict__ x, float* __restrict__ out) {
    // --- Build the 64x20 sinusoid table in LDS (5 KB), 5 trig evals/thread ---
    __shared__ __align__(16) float pe[kC * kT];
    for (int i = threadIdx.x; i < kC * kT; i += kThreads) {
        int c = i / kT;
        int t = i - c * kT;
        // inv_freq = 10000^(-2*(c/2)/C) = exp(-(2*(c/2)/C) * ln(10000))
        float e        = (-2.0f * (float)(c >> 1) / (float)kC) * 9.2103403719761836f;
        float inv_freq = expf(e);
        float ang      = inv_freq * (float)t;
        pe[i] = (c & 1) ? cosf(ang) : sinf(ang);
    }
    __syncthreads();

    // gridDim.y enumerates the 256 (b,c) slabs; c is the low 6 bits.
    const int    c         = (int)(blockIdx.y & (kC - 1));
    const float* pe_c      = &pe[c * kT];
    const size_t slab_base = (size_t)blockIdx.y * (size_t)kSlabVec4;

    const v4f* __restrict__ xv = (const v4f*)x   + slab_base;
    v4f*       __restrict__ ov = (v4f*)out       + slab_base;

    int i4 = (int)blockIdx.x * (kThreads * kVecPerThr) + (int)threadIdx.x;
#pragma unroll
    for (int k = 0; k < kVecPerThr; ++k, i4 += kThreads) {
        // float4 at element offset 4*i4: t0 = (4*i4) % 20 = (i4 % 5) * 4.
        // t0 in {0,4,8,12,16} -> never crosses the t-period; LDS read is
        // 16B aligned (byte offset c*80 + t0*4) -> single ds_load_b128.
        const int t0 = (i4 % 5) * 4;
        v4f v = __builtin_nontemporal_load(xv + i4);         // global_load_b128 th:NT
        v4f p = *(const v4f*)(pe_c + t0);                    // ds_load_b128
        v += p;
        __builtin_nontemporal_store(v, ov + i4);             // global_store_b128 th:NT
    }
}

extern "C" void kernel_launch(void* const* d_in, const int* in_sizes, int n_in,
                              void* d_out, int out_size, void* d_ws, size_t ws_size,
                              hipStream_t stream) {
    const float* x   = (const float*)d_in[0];   // [4,64,128,128,20] fp32
    // d_in[1] is the scalar t (==20), baked into the tiling constants.
    float*       out = (float*)d_out;

    dim3 grid(kBlocksX, kB * kC, 1);            // 80 x 256 blocks
    pe_add_kernel<<<grid, kThreads, 0, stream>>>(x, out);
}